// GaussianCircular_8967891714434
// MI455X (gfx1250) — compile-verified
//
#include <hip/hip_runtime.h>
#include <math.h>

typedef __attribute__((ext_vector_type(2))) float v2f;
typedef __attribute__((ext_vector_type(8))) float v8f;
typedef int v4i __attribute__((ext_vector_type(4)));

#define HALF       7
#define IMG        1024
#define TILE_X     128          // per workgroup: 8 wave32 x 16 cols
#define TILE_Y     16
#define XPAD       8            // patch col c <-> gx = tx0 - 8 + c (16B-aligned origin)
#define PATCH_ROWS 32           // rows 0..29 real (ty0-7..ty0+22), 30/31 filler
#define PATCH_COLS 148          // covers gx up to tx0+139; 592 B stride (16B multiple)
#define TMPT_STRIDE 36          // transposed tmp [128][36]; 144 B stride (16B multiple)

#if defined(__gfx1250__) && __has_builtin(__builtin_amdgcn_global_load_async_to_lds_b128)
#define HAVE_ASYNC 1
typedef __attribute__((address_space(1))) v4i* g_v4i_p;   // global int4*
typedef __attribute__((address_space(3))) v4i* l_v4i_p;   // LDS int4*
#endif

// D = A(16x4) * B(4x16) + C, f32 WMMA, wave32.
static __device__ __forceinline__ v8f wmma4(v2f a, v2f b, v8f c) {
    return __builtin_amdgcn_wmma_f32_16x16x4_f32(
        false, a, false, b, (short)0, c, false, false);
}

// g(d) = exp(-d^2 / (2 sigma^2)) inside the 15-tap band, exactly 0 outside.
static __device__ __forceinline__ float bandw(int d, float neg_inv2s2) {
    float w = __expf((float)(d * d) * neg_inv2s2);
    return ((unsigned)(d + HALF) <= 14u) ? w : 0.0f;
}

__global__ __launch_bounds__(256)
void gauss15_sep_wmma(const float* __restrict__ in,
                      const float* __restrict__ sigma_p,
                      const float* __restrict__ gain_p,
                      float* __restrict__ out)
{
    __shared__ float s_in[PATCH_ROWS][PATCH_COLS];
    __shared__ float s_tmpt[TILE_X][TMPT_STRIDE];   // [col][row] (transposed tmp)

    const int tid  = threadIdx.x;
    const int lane = tid & 31;
    const int wave = tid >> 5;

    const int tx0 = blockIdx.x * TILE_X;
    const int ty0 = blockIdx.y * TILE_Y;
    const size_t ioff = (size_t)blockIdx.z * IMG * IMG;
    const float* __restrict__ inp  = in  + ioff;
    float* __restrict__ outp       = out + ioff;

    // Block-uniform: whole 32x148 patch in-bounds -> async B128 path is legal.
    const bool interior = (ty0 >= HALF) && (ty0 + 25 <= IMG) &&
                          (tx0 >= XPAD) && (tx0 - XPAD + PATCH_COLS <= IMG);

#ifdef HAVE_ASYNC
    if (interior) {
        // 32 rows x 37 float4 = 1184 aligned 16B transfers, memory -> LDS direct.
        const int NV4 = PATCH_ROWS * (PATCH_COLS / 4);
        for (int i = tid; i < NV4; i += 256) {
            int r  = i / (PATCH_COLS / 4);
            int c4 = i - r * (PATCH_COLS / 4);
            const float* g = inp + (size_t)(ty0 - HALF + r) * IMG + (tx0 - XPAD) + c4 * 4;
            unsigned loff = (unsigned)(uintptr_t)&s_in[r][c4 * 4];
            __builtin_amdgcn_global_load_async_to_lds_b128(
                (g_v4i_p)(uintptr_t)g, (l_v4i_p)loff, 0, 0);
        }
#if __has_builtin(__builtin_amdgcn_s_wait_asynccnt)
        __builtin_amdgcn_s_wait_asynccnt(0);
#else
        asm volatile("s_wait_asynccnt 0" ::: "memory");
#endif
    } else
#endif
    {
        // Border tiles: bounds-checked loads implement SAME zero padding.
        for (int i = tid; i < PATCH_ROWS * PATCH_COLS; i += 256) {
            int r = i / PATCH_COLS;
            int c = i - r * PATCH_COLS;
            int gy = ty0 - HALF + r;
            int gx = tx0 - XPAD + c;
            float v = 0.0f;
            if ((unsigned)gy < IMG && (unsigned)gx < IMG)
                v = inp[(size_t)gy * IMG + gx];
            s_in[r][c] = v;
        }
    }

    // Band weights in registers; shared between both passes (no LDS table,
    // no exec-divergent selects around the WMMAs).
    const int n     = lane & 15;
    const int khalf = (lane < 16) ? 0 : 2;
    const int x0    = wave * 16;

    float s  = fabsf(sigma_p[0]);
    float ni = -1.0f / (2.0f * s * s);
    float w8[8], w7[8], w6[8];
    #pragma unroll
    for (int kc = 0; kc < 8; ++kc) {
        int db = kc * 4 + khalf - n;
        w8[kc] = bandw(db - 8, ni);   // pass1 b.x : g(k0   - x - 8)
        w7[kc] = bandw(db - 7, ni);   // pass1 b.y / pass2 a.x
        w6[kc] = bandw(db - 6, ni);   // pass2 a.y : g(k0+1 - y - 7)
    }

    __syncthreads();

    // ---- Pass 1 (horizontal): Tmp(16x16) = In(16x32) x W(32x16), 2 row blocks.
    #pragma unroll
    for (int rb = 0; rb < 2; ++rb) {
        const int rbase = rb * 16;
        v8f acc = {};
        #pragma unroll
        for (int kc = 0; kc < 8; ++kc) {
            const int k0 = kc * 4 + khalf;
            v2f a, b;
            a.x = s_in[rbase + n][x0 + k0];       // aligned ds_load_b64
            a.y = s_in[rbase + n][x0 + k0 + 1];
            b.x = w8[kc];
            b.y = w7[kc];
            acc = wmma4(a, b, acc);
        }
        // D rows are contiguous per lane in the transposed tmp -> 2x b128 stores.
        const int col = x0 + n;
        const int r0  = rbase + ((lane < 16) ? 0 : 8);
        *(float4*)&s_tmpt[col][r0]     = make_float4(acc[0], acc[1], acc[2], acc[3]);
        *(float4*)&s_tmpt[col][r0 + 4] = make_float4(acc[4], acc[5], acc[6], acc[7]);
    }
    __syncthreads();

    // ---- Pass 2 (vertical): Out(16x16) = Gv(16x32) x Tmp(32x16).
    const float gain = gain_p[0];
    const int col = x0 + n;
    v8f acc = {};
    #pragma unroll
    for (int kc = 0; kc < 8; ++kc) {
        const int k0 = kc * 4 + khalf;
        v2f a, b;
        a.x = w7[kc];
        a.y = w6[kc];
        b.x = s_tmpt[col][k0];                    // aligned ds_load_b64
        b.y = s_tmpt[col][k0 + 1];
        acc = wmma4(a, b, acc);
    }
    #pragma unroll
    for (int j = 0; j < 8; ++j) {
        int row = (lane < 16) ? j : j + 8;
        outp[(size_t)(ty0 + row) * IMG + tx0 + col] = gain * acc[j];
    }
}

extern "C" void kernel_launch(void* const* d_in, const int* in_sizes, int n_in,
                              void* d_out, int out_size, void* d_ws, size_t ws_size,
                              hipStream_t stream)
{
    (void)in_sizes; (void)n_in; (void)d_ws; (void)ws_size; (void)out_size;
    const float* x     = (const float*)d_in[0];
    const float* sigma = (const float*)d_in[1];
    const float* gain  = (const float*)d_in[2];
    float* outp        = (float*)d_out;

    dim3 grid(IMG / TILE_X, IMG / TILE_Y, 16);   // 8 x 64 x 16
    dim3 block(256);
    gauss15_sep_wmma<<<grid, block, 0, stream>>>(x, sigma, gain, outp);
}